// GraphNeuralNetwork_64192581206328
// MI455X (gfx1250) — compile-verified
//
#include <hip/hip_runtime.h>
#include <hip/hip_bf16.h>

typedef __attribute__((ext_vector_type(2))) float v2f;
typedef __attribute__((ext_vector_type(8))) float v8f;

// ---------------------------------------------------------------------------
// Graph normalization precompute (shared across all 3 layers)
// ---------------------------------------------------------------------------
__global__ void init_deg_kernel(float* __restrict__ deg, int n) {
    int i = blockIdx.x * blockDim.x + threadIdx.x;
    if (i < n) deg[i] = 1.0f;  // self-loop weight
}

__global__ void deg_accum_kernel(float* __restrict__ deg, const int* __restrict__ cols,
                                 const float* __restrict__ w, int e) {
    int i = blockIdx.x * blockDim.x + threadIdx.x;
    if (i < e) atomicAdd(&deg[cols[i]], w[i]);
}

__global__ void dinv_kernel(float* __restrict__ deg, int n) {
    int i = blockIdx.x * blockDim.x + threadIdx.x;
    if (i < n) {
        float d = deg[i];
        deg[i] = (d > 0.0f) ? rsqrtf(d) : 0.0f;  // in-place deg -> dinv
    }
}

__global__ void coef_kernel(float* __restrict__ coef, const int* __restrict__ rows,
                            const int* __restrict__ cols, const float* __restrict__ w,
                            const float* __restrict__ dinv, int e, int n) {
    int i = blockIdx.x * blockDim.x + threadIdx.x;
    if (i < e) {
        coef[i] = dinv[rows[i]] * w[i] * dinv[cols[i]];
    } else if (i < e + n) {
        float d = dinv[i - e];
        coef[i] = d * d;  // self-loop, weight 1.0
    }
}

// ---------------------------------------------------------------------------
// Dense GEMM: Y[N,F] = X[N,K] @ W[K,F] via V_WMMA_F32_16X16X4_F32.
// One wave -> 16-row strip x full F. W staged in LDS in a k-pair-interleaved,
// bank-conflict-free layout so each B fragment is a single ds_load_b64.
// ---------------------------------------------------------------------------
template <int K, int F>
__global__ void gemm_wmma_kernel(const float* __restrict__ X, const float* __restrict__ W,
                                 float* __restrict__ Y, int nRows) {
    constexpr int KSLAB = 64;
    constexpr int NPAIR = KSLAB / 2;       // 32 k-pair rows per slab
    constexpr int PAD = 32;                // floats: pair stride = 128B mod 256B
    constexpr int STRIDE = 2 * F + PAD;    // floats per pair-row in LDS
    constexpr int NT = F / 16;             // output column tiles per wave
    __shared__ float ldsW[NPAIR * STRIDE];

    const int waveInBlk = threadIdx.x >> 5;
    const int lane = threadIdx.x & 31;
    const int tileM = blockIdx.x * 8 + waveInBlk;
    const int row0 = tileM * 16;
    const bool active = (row0 < nRows);

    const int m16 = lane & 15;
    const int khalf = (lane >> 4) * 2;  // A/B fragment K sub-offset: 0 or 2
    const long mrow = (long)(row0 + m16);

    v8f acc[NT] = {};

    for (int k0 = 0; k0 < K; k0 += KSLAB) {
        __syncthreads();
        // cooperative interleaved fill: lds[p*STRIDE + c*2 + {0,1}] = W[k0+2p][c], W[k0+2p+1][c]
        for (int i = threadIdx.x; i < NPAIR * (F / 2); i += blockDim.x) {
            const int p = i / (F / 2);
            const int cq = i % (F / 2);
            const long gbase = (long)(k0 + 2 * p) * F + cq * 2;
            const float2 g0 = *(const float2*)(W + gbase);        // row 2p,   cols cq*2..+1
            const float2 g1 = *(const float2*)(W + gbase + F);    // row 2p+1, cols cq*2..+1
            *(float4*)(&ldsW[p * STRIDE + cq * 4]) = make_float4(g0.x, g1.x, g0.y, g1.y);
        }
        __syncthreads();

        if (active) {
            for (int kk = 0; kk < KSLAB; kk += 4) {
                // A fragment: 16x4 fp32 (ISA 7.12.2 32-bit A layout)
                v2f a = *(const v2f*)(X + mrow * K + k0 + kk + khalf);
                const int pbase = (kk + khalf) >> 1;  // kk+khalf is even
#pragma unroll
                for (int t = 0; t < NT; ++t) {
                    const int col = t * 16 + m16;
                    // single aligned ds_load_b64: {W[k][col], W[k+1][col]}
                    v2f b = *(const v2f*)(&ldsW[pbase * STRIDE + col * 2]);
                    acc[t] = __builtin_amdgcn_wmma_f32_16x16x4_f32(
                        false, a, false, b, (short)0, acc[t], false, false);
                }
            }
        }
    }

    if (active) {
        // C/D layout: VGPR i -> M = i + 8*(lane>=16), N = lane&15
        const int rOut = row0 + (lane >> 4) * 8;
#pragma unroll
        for (int t = 0; t < NT; ++t) {
            const int c = t * 16 + m16;
#pragma unroll
            for (int i = 0; i < 8; ++i)
                Y[(long)(rOut + i) * F + c] = acc[t][i];
        }
    }
}

// ---------------------------------------------------------------------------
// Sparse aggregation: out[cols[e]] += coef[e] * h[rows[e]]   (one wave / edge)
// ---------------------------------------------------------------------------
template <int F>
__global__ void agg_kernel(const float* __restrict__ h, const int* __restrict__ rows,
                           const int* __restrict__ cols, const float* __restrict__ coef,
                           float* __restrict__ out, int e, int totalEdges) {
    const int wave = (int)((blockIdx.x * (long)blockDim.x + threadIdx.x) >> 5);
    const int lane = threadIdx.x & 31;
    if (wave >= totalEdges) return;

    int r, c;
    if (wave < e) { r = rows[wave]; c = cols[wave]; }
    else          { r = c = wave - e; }
    const float cf = coef[wave];

    if (F == 128) {
        float4 v = ((const float4*)(h + (long)r * F))[lane];
        float* op = out + (long)c * F + lane * 4;
        atomicAdd(op + 0, cf * v.x);
        atomicAdd(op + 1, cf * v.y);
        atomicAdd(op + 2, cf * v.z);
        atomicAdd(op + 3, cf * v.w);
    } else {  // F == 64
        float2 v = ((const float2*)(h + (long)r * F))[lane];
        float* op = out + (long)c * F + lane * 2;
        atomicAdd(op + 0, cf * v.x);
        atomicAdd(op + 1, cf * v.y);
    }
}

// ---------------------------------------------------------------------------
// BatchNorm (128 features): two-level reduction, then fused scale/shift + ReLU
// ---------------------------------------------------------------------------
__global__ void bn_stats_kernel(const float* __restrict__ x, float* __restrict__ gsum,
                                float* __restrict__ gsq, int nRows) {
    __shared__ float s0[256];
    __shared__ float s1[256];
    const int f = threadIdx.x & 127;
    const int sub = threadIdx.x >> 7;  // 0 or 1
    const int rBeg = blockIdx.x * 256 + sub;
    int rEnd = blockIdx.x * 256 + 256;
    if (rEnd > nRows) rEnd = nRows;
    float s = 0.f, q = 0.f;
    for (int r = rBeg; r < rEnd; r += 2) {
        float v = x[(long)r * 128 + f];
        s += v;
        q += v * v;
    }
    s0[threadIdx.x] = s;
    s1[threadIdx.x] = q;
    __syncthreads();
    if (sub == 0) {
        s += s0[threadIdx.x + 128];
        q += s1[threadIdx.x + 128];
        atomicAdd(&gsum[f], s);
        atomicAdd(&gsq[f], q);
    }
}

__global__ void bn_finalize_kernel(const float* __restrict__ gsum, const float* __restrict__ gsq,
                                   const float* __restrict__ g, const float* __restrict__ be,
                                   float* __restrict__ scale, float* __restrict__ shift,
                                   float invN) {
    int f = threadIdx.x;  // 128 threads
    float mu = gsum[f] * invN;
    float var = gsq[f] * invN - mu * mu;
    float sc = rsqrtf(var + 1e-5f) * g[f];
    scale[f] = sc;
    shift[f] = be[f] - mu * sc;
}

__global__ void bn_apply_relu_kernel(float* __restrict__ x, const float* __restrict__ scale,
                                     const float* __restrict__ shift, long n4) {
    long i = blockIdx.x * (long)blockDim.x + threadIdx.x;
    if (i >= n4) return;
    const int f0 = (int)((i * 4) & 127);
    float4 v = ((float4*)x)[i];
    v.x = fmaxf(0.f, v.x * scale[f0 + 0] + shift[f0 + 0]);
    v.y = fmaxf(0.f, v.y * scale[f0 + 1] + shift[f0 + 1]);
    v.z = fmaxf(0.f, v.z * scale[f0 + 2] + shift[f0 + 2]);
    v.w = fmaxf(0.f, v.w * scale[f0 + 3] + shift[f0 + 3]);
    ((float4*)x)[i] = v;
}

// ---------------------------------------------------------------------------
// Utility kernels
// ---------------------------------------------------------------------------
__global__ void zero_kernel(float* __restrict__ p, long n4) {
    long i = blockIdx.x * (long)blockDim.x + threadIdx.x;
    if (i < n4) ((float4*)p)[i] = make_float4(0.f, 0.f, 0.f, 0.f);
}

__global__ void init_out_bias_kernel(float* __restrict__ out, const float* __restrict__ b3,
                                     long n) {
    long i = blockIdx.x * (long)blockDim.x + threadIdx.x;
    if (i < n) out[i] = b3[i & 63];
}

// ---------------------------------------------------------------------------
// Host-side orchestration (all on `stream`, graph-capture safe)
// ---------------------------------------------------------------------------
static inline size_t alignUp(size_t v, size_t a) { return (v + a - 1) & ~(a - 1); }

extern "C" void kernel_launch(void* const* d_in, const int* in_sizes, int n_in,
                              void* d_out, int out_size, void* d_ws, size_t ws_size,
                              hipStream_t stream) {
    const float* x  = (const float*)d_in[0];   // [N,256]
    const int*   ei = (const int*)d_in[1];     // [2,E] flat
    const float* ew = (const float*)d_in[2];   // [E]
    const float* W1 = (const float*)d_in[3];
    // d_in[4] = b1 (absorbed by BN)
    const float* g1  = (const float*)d_in[5];
    const float* be1 = (const float*)d_in[6];
    const float* W2  = (const float*)d_in[7];
    // d_in[8] = b2 (absorbed by BN)
    const float* g2  = (const float*)d_in[9];
    const float* be2 = (const float*)d_in[10];
    const float* W3  = (const float*)d_in[11];
    const float* b3  = (const float*)d_in[12];

    const int N = in_sizes[0] / 256;
    const int E = in_sizes[2];
    const int T = E + N;  // total edges incl. self-loops
    const int* rows = ei;
    const int* cols = ei + E;

    // workspace carve-up
    char* base = (char*)d_ws;
    size_t off = 0;
    float* dinv = (float*)(base + off); off = alignUp(off + (size_t)N * 4, 256);
    float* coef = (float*)(base + off); off = alignUp(off + (size_t)T * 4, 256);
    float* bufA = (float*)(base + off); off = alignUp(off + (size_t)N * 128 * 4, 256);
    float* bufB = (float*)(base + off); off = alignUp(off + (size_t)N * 128 * 4, 256);
    float* gsum  = (float*)(base + off); off = alignUp(off + 128 * 4, 256);
    float* gsq   = (float*)(base + off); off = alignUp(off + 128 * 4, 256);
    float* scale = (float*)(base + off); off = alignUp(off + 128 * 4, 256);
    float* shift = (float*)(base + off); off = alignUp(off + 128 * 4, 256);
    (void)ws_size;

    const int BT = 256;
    const int mTiles = (N + 15) / 16;
    const int gemmBlocks = (mTiles + 7) / 8;
    const int aggBlocks = (T + 7) / 8;              // 8 waves (edges) per block
    const long n4_128 = (long)N * 128 / 4;
    const int elwBlocks128 = (int)((n4_128 + BT - 1) / BT);
    const int bnBlocks = (N + 255) / 256;
    const float invN = 1.0f / (float)N;

    // --- graph normalization (once, reused by all layers) ---
    init_deg_kernel<<<(N + BT - 1) / BT, BT, 0, stream>>>(dinv, N);
    deg_accum_kernel<<<(E + BT - 1) / BT, BT, 0, stream>>>(dinv, cols, ew, E);
    dinv_kernel<<<(N + BT - 1) / BT, BT, 0, stream>>>(dinv, N);
    coef_kernel<<<(T + BT - 1) / BT, BT, 0, stream>>>(coef, rows, cols, ew, dinv, E, N);

    // --- layer 1: x @ W1 -> agg -> BN+ReLU ---
    gemm_wmma_kernel<256, 128><<<gemmBlocks, BT, 0, stream>>>(x, W1, bufA, N);
    zero_kernel<<<elwBlocks128, BT, 0, stream>>>(bufB, n4_128);
    agg_kernel<128><<<aggBlocks, BT, 0, stream>>>(bufA, rows, cols, coef, bufB, E, T);
    zero_kernel<<<1, 64, 0, stream>>>(gsum, 32);
    zero_kernel<<<1, 64, 0, stream>>>(gsq, 32);
    bn_stats_kernel<<<bnBlocks, BT, 0, stream>>>(bufB, gsum, gsq, N);
    bn_finalize_kernel<<<1, 128, 0, stream>>>(gsum, gsq, g1, be1, scale, shift, invN);
    bn_apply_relu_kernel<<<elwBlocks128, BT, 0, stream>>>(bufB, scale, shift, n4_128);

    // --- layer 2 ---
    gemm_wmma_kernel<128, 128><<<gemmBlocks, BT, 0, stream>>>(bufB, W2, bufA, N);
    zero_kernel<<<elwBlocks128, BT, 0, stream>>>(bufB, n4_128);
    agg_kernel<128><<<aggBlocks, BT, 0, stream>>>(bufA, rows, cols, coef, bufB, E, T);
    zero_kernel<<<1, 64, 0, stream>>>(gsum, 32);
    zero_kernel<<<1, 64, 0, stream>>>(gsq, 32);
    bn_stats_kernel<<<bnBlocks, BT, 0, stream>>>(bufB, gsum, gsq, N);
    bn_finalize_kernel<<<1, 128, 0, stream>>>(gsum, gsq, g2, be2, scale, shift, invN);
    bn_apply_relu_kernel<<<elwBlocks128, BT, 0, stream>>>(bufB, scale, shift, n4_128);

    // --- layer 3: output initialized with b3, aggregation adds on top ---
    gemm_wmma_kernel<128, 64><<<gemmBlocks, BT, 0, stream>>>(bufB, W3, bufA, N);
    const long nOut = (long)N * 64;
    init_out_bias_kernel<<<(int)((nOut + BT - 1) / BT), BT, 0, stream>>>((float*)d_out, b3, nOut);
    agg_kernel<64><<<aggBlocks, BT, 0, stream>>>(bufA, rows, cols, coef, (float*)d_out, E, T);
}